// SymphoniesLayer_51951924413035
// MI455X (gfx1250) — compile-verified
//
#include <hip/hip_runtime.h>

// ---------------------------------------------------------------------------
// SymphoniesLayer for MI455X (gfx1250, wave32).
//
// Roofline: dominant work = fp32 GEMMs (2x 262144x128x128 value projections,
// 32768-row transformer/FFN GEMMs, ~45 GFLOP total) -> routed through
// V_WMMA_F32_16X16X4_F32 (exact fp32 matrix pipe). GEMM wave tile is
// 16x64 (4 independent WMMA accumulator chains) over a 64x128 block tile;
// B is staged pair-interleaved in LDS so every fragment is one ds_load_b64
// (no repack movs); A staging uses GLOBAL_LOAD_ASYNC_TO_LDS (ASYNCcnt path).
// Attention (Nk=100), LN and deformable sampling are bandwidth/latency bound
// -> scalar wave32 kernels with shfl reductions. 134MB scene copy + fov
// scatter via LDS-tiled transposes (coalesced both sides).
// ---------------------------------------------------------------------------

typedef __attribute__((ext_vector_type(2))) float v2f;
typedef __attribute__((ext_vector_type(8))) float v8f;

#define NVOX  (128 * 128 * 16)
#define NFOV  32768
#define NINST 100

// ---------------------------------------------------------------------------
// WMMA f32 16x16x4 (D = A(16x4) * B(4x16) + C), wave32. (codegen-confirmed)
// ---------------------------------------------------------------------------
#if __has_builtin(__builtin_amdgcn_wmma_f32_16x16x4_f32)
#define WMMA_F32_K4(a, b, c) \
  __builtin_amdgcn_wmma_f32_16x16x4_f32(false, (a), false, (b), (short)0, (c), false, false)
#else
static __device__ inline v8f wmma_f32_k4_emu(v2f a, v2f b, v8f c) {
  int lane = threadIdx.x & 31;
  int rr = lane & 15, hf = lane >> 4;
#pragma unroll
  for (int j = 0; j < 8; j++) {
    int m = (hf ? 8 : 0) + j;
    float s = c[j];
#pragma unroll
    for (int k = 0; k < 4; k++) {
      int asrc = (m & 15) + ((k >> 1) << 4);
      int bsrc = rr + ((k >> 1) << 4);
      float av = __shfl((k & 1) ? a.y : a.x, asrc, 32);
      float bv = __shfl((k & 1) ? b.y : b.x, bsrc, 32);
      s += av * bv;
    }
    c[j] = s;
  }
  return c;
}
#define WMMA_F32_K4(a, b, c) wmma_f32_k4_emu((a), (b), (c))
#endif

// Async global->LDS staging (ASYNCcnt path). Probe-confirmed present on this
// toolchain; parameter types from clang diagnostics: b32 -> (int*, ..., Ii, Ii),
// b128 -> (vector_size(16) int*, ..., Ii, Ii), generic address spaces.
#if __has_builtin(__builtin_amdgcn_global_load_async_to_lds_b128) && \
    __has_builtin(__builtin_amdgcn_global_load_async_to_lds_b32) &&  \
    __has_builtin(__builtin_amdgcn_s_wait_asynccnt)
#define HAVE_ASYNC_LDS 1
typedef int async_b32_t;
typedef __attribute__((__vector_size__(4 * sizeof(int)))) int async_b128_t;
#else
#define HAVE_ASYNC_LDS 0
#endif

#define AS_PAD    132  // A row stride (floats): 16B-aligned rows, conflict-free frags
#define BS_STRIDE 288  // B k-pair row stride (floats): 256 data + 32 pad (bank split)

// ---------------------------------------------------------------------------
// Generic GEMM: out[M,N] = epi(A[M,K] @ W[K,N] + bias, res)
//   a_trans=0: A row-major, row stride lda (=K)
//   a_trans=1: A channel-major: A[r][k] = A[k*lda + r]
//   epi: 0 = bias, 1 = bias+ReLU, 2 = bias+residual(res[M,N])
// Block: 256 threads (8 waves). Block tile 64(M) x 128(N); wave (wm 0..3,
// wn 0..1) computes 16 rows x 64 cols = 4 x (16x16) accumulator chains.
// B staged as {k,k+1} interleaved pairs -> every fragment is one ds_load_b64.
// ---------------------------------------------------------------------------
__global__ __launch_bounds__(256) void gemm_wmma_kernel(
    const float* __restrict__ A, int lda, int a_trans,
    const float* __restrict__ W, const float* __restrict__ bias,
    const float* __restrict__ res, float* __restrict__ out,
    int M, int N, int K, int epi) {
  extern __shared__ float smem[];
  float* As = smem;                 // [64][AS_PAD]
  float* Bs = smem + 64 * AS_PAD;   // [64 k-pairs][BS_STRIDE] interleaved

  const int tid = threadIdx.x;
  const int wave = tid >> 5, lane = tid & 31;
  const int wm = wave >> 1;  // 0..3 : 16-row subtile
  const int wn = wave & 1;   // 0..1 : 64-col subtile
  const int m_blk = blockIdx.x * 64;
  const int n_blk = blockIdx.y * 128;
  const int rr = lane & 15;
  const int kb = (lane >> 4) << 1;  // 0 or 2

  const v8f vzero = {0, 0, 0, 0, 0, 0, 0, 0};
  v8f acc[4];
  acc[0] = vzero; acc[1] = vzero; acc[2] = vzero; acc[3] = vzero;

  for (int k0 = 0; k0 < K; k0 += 128) {
    // ---- stage A chunk (64 rows x 128 k) ----------------------------------
#if HAVE_ASYNC_LDS
    if (!a_trans) {
      for (int idx = tid; idx < 64 * 32; idx += 256) {
        int r = idx >> 5, c4 = (idx & 31) * 4;
        int gr = m_blk + r;
        float* dst = As + r * AS_PAD + c4;
        if (gr < M) {
          __builtin_amdgcn_global_load_async_to_lds_b128(
              (async_b128_t*)(A + (size_t)gr * lda + k0 + c4),
              (async_b128_t*)dst, 0, 0);
        } else {
          dst[0] = 0.f; dst[1] = 0.f; dst[2] = 0.f; dst[3] = 0.f;
        }
      }
    } else {
      for (int idx = tid; idx < 64 * 128; idx += 256) {
        int r = idx & 63, c = idx >> 6;
        int gr = m_blk + r;
        float* dst = As + r * AS_PAD + c;
        if (gr < M) {
          __builtin_amdgcn_global_load_async_to_lds_b32(
              (async_b32_t*)(A + (size_t)(k0 + c) * lda + gr),
              (async_b32_t*)dst, 0, 0);
        } else {
          dst[0] = 0.f;
        }
      }
    }
#else
    if (!a_trans) {
      for (int idx = tid; idx < 64 * 128; idx += 256) {
        int r = idx >> 7, c = idx & 127;
        int gr = m_blk + r;
        As[r * AS_PAD + c] = (gr < M) ? A[(size_t)gr * lda + k0 + c] : 0.f;
      }
    } else {
      for (int idx = tid; idx < 64 * 128; idx += 256) {
        int r = idx & 63, c = idx >> 6;
        int gr = m_blk + r;
        As[r * AS_PAD + c] = (gr < M) ? A[(size_t)(k0 + c) * lda + gr] : 0.f;
      }
    }
#endif
    // ---- stage B chunk (128 k x 128 n), {k,k+1} interleaved ---------------
    // All callers use N % 16 == 0, so a 4-col quad is entirely in or out.
    for (int idx = tid; idx < 64 * 32; idx += 256) {
      int p = idx >> 5, cq = (idx & 31) * 4;
      int gc = n_blk + cq;
      float4 r0 = make_float4(0.f, 0.f, 0.f, 0.f);
      float4 r1 = make_float4(0.f, 0.f, 0.f, 0.f);
      if (gc < N) {
        r0 = *(const float4*)(W + (size_t)(k0 + 2 * p) * N + gc);
        r1 = *(const float4*)(W + (size_t)(k0 + 2 * p + 1) * N + gc);
      }
      float* dst = Bs + p * BS_STRIDE + cq * 2;
      ((float4*)dst)[0] = make_float4(r0.x, r1.x, r0.y, r1.y);
      ((float4*)dst)[1] = make_float4(r0.z, r1.z, r0.w, r1.w);
    }
#if HAVE_ASYNC_LDS
    __builtin_amdgcn_s_wait_asynccnt(0);
#endif
    __syncthreads();

    const float* ap = As + (wm * 16 + rr) * AS_PAD;
    const int cb = wn * 64 + rr;
#pragma unroll
    for (int kk = 0; kk < 128; kk += 4) {
      v2f a;
      a.x = ap[kk + kb];
      a.y = ap[kk + kb + 1];
      const v2f* bp = (const v2f*)(Bs + ((kk + kb) >> 1) * BS_STRIDE);
      v2f b0 = bp[cb];
      v2f b1 = bp[cb + 16];
      v2f b2 = bp[cb + 32];
      v2f b3 = bp[cb + 48];
      acc[0] = WMMA_F32_K4(a, b0, acc[0]);
      acc[1] = WMMA_F32_K4(a, b1, acc[1]);
      acc[2] = WMMA_F32_K4(a, b2, acc[2]);
      acc[3] = WMMA_F32_K4(a, b3, acc[3]);
    }
    __syncthreads();
  }

  const int hf = lane >> 4;
#pragma unroll
  for (int j = 0; j < 8; j++) {
    int grow = m_blk + wm * 16 + hf * 8 + j;
    if (grow >= M) continue;
    size_t ro = (size_t)grow * N;
#pragma unroll
    for (int t = 0; t < 4; t++) {
      int gc = n_blk + wn * 64 + t * 16 + rr;
      if (gc < N) {
        float v = acc[t][j] + bias[gc];
        if (epi == 1) v = fmaxf(v, 0.f);
        else if (epi == 2) v += res[ro + gc];
        out[ro + gc] = v;
      }
    }
  }
}

// ---------------------------------------------------------------------------
// LayerNorm over rows of width 128. One wave per row (4 floats/lane).
// ---------------------------------------------------------------------------
__device__ inline float wave_sum(float v) {
#pragma unroll
  for (int o = 16; o > 0; o >>= 1) v += __shfl_xor(v, o, 32);
  return v;
}
__device__ inline float wave_max(float v) {
#pragma unroll
  for (int o = 16; o > 0; o >>= 1) v = fmaxf(v, __shfl_xor(v, o, 32));
  return v;
}

__global__ __launch_bounds__(256) void layernorm_kernel(
    const float* __restrict__ x, const float* __restrict__ g,
    const float* __restrict__ b, float* __restrict__ out, int M) {
  int row = blockIdx.x * 8 + (threadIdx.x >> 5);
  int lane = threadIdx.x & 31;
  if (row >= M) return;
  const float4 v = ((const float4*)(x + (size_t)row * 128))[lane];
  float mean = wave_sum(v.x + v.y + v.z + v.w) * (1.f / 128.f);
  float dx = v.x - mean, dy = v.y - mean, dz = v.z - mean, dw = v.w - mean;
  float var = wave_sum(dx * dx + dy * dy + dz * dz + dw * dw) * (1.f / 128.f);
  float rs = rsqrtf(var + 1e-5f);
  const float4 gg = ((const float4*)g)[lane];
  const float4 bb = ((const float4*)b)[lane];
  float4 o;
  o.x = dx * rs * gg.x + bb.x;
  o.y = dy * rs * gg.y + bb.y;
  o.z = dz * rs * gg.z + bb.z;
  o.w = dw * rs * gg.w + bb.w;
  ((float4*)(out + (size_t)row * 128))[lane] = o;
}

// ---------------------------------------------------------------------------
// MHA core: per (query, head) one wave; Nk <= 128 (here 100). HD=16, NH=8.
// qh/kh/vh: [N][128] laid out as [N][h*16+d]. out: [Nq][128].
// ---------------------------------------------------------------------------
__global__ __launch_bounds__(256) void mha_attn_kernel(
    const float* __restrict__ qh, const float* __restrict__ kh,
    const float* __restrict__ vh, float* __restrict__ out, int Nq, int Nk) {
  int q = blockIdx.x;
  int h = threadIdx.x >> 5, lane = threadIdx.x & 31;
  const float* qp = qh + (size_t)q * 128 + h * 16;
  float ql[16];
#pragma unroll
  for (int d = 0; d < 16; d++) ql[d] = qp[d];

  float lg[4];
#pragma unroll
  for (int j = 0; j < 4; j++) {
    int k = lane + 32 * j;
    float s = -1e30f;
    if (k < Nk) {
      const float* kp = kh + (size_t)k * 128 + h * 16;
      float acc = 0.f;
#pragma unroll
      for (int d = 0; d < 16; d++) acc += ql[d] * kp[d];
      s = acc * 0.25f; // 1/sqrt(HD=16)
    }
    lg[j] = s;
  }
  float m = wave_max(fmaxf(fmaxf(lg[0], lg[1]), fmaxf(lg[2], lg[3])));

  float po[16];
#pragma unroll
  for (int d = 0; d < 16; d++) po[d] = 0.f;
  float esum = 0.f;
#pragma unroll
  for (int j = 0; j < 4; j++) {
    int k = lane + 32 * j;
    if (k < Nk) {
      float e = __expf(lg[j] - m);
      esum += e;
      const float* vp = vh + (size_t)k * 128 + h * 16;
#pragma unroll
      for (int d = 0; d < 16; d++) po[d] += e * vp[d];
    }
  }
  float inv = 1.f / wave_sum(esum);

  float oval = 0.f;
#pragma unroll
  for (int d = 0; d < 16; d++) {
    float r = wave_sum(po[d]);
    if (lane == d) oval = r;
  }
  if (lane < 16) out[(size_t)q * 128 + h * 16 + lane] = oval * inv;
}

// ---------------------------------------------------------------------------
// Deformable sampling, 3D (single level = GRID 128x128x16, 8 points).
// v: [NVOX][128], ref: [Nq][3], off: [Nq][192], aw: [Nq][64]. One thread per
// (q, head).
// ---------------------------------------------------------------------------
__global__ __launch_bounds__(256) void deform_sample_3d_kernel(
    const float* __restrict__ v, const float* __restrict__ ref,
    const float* __restrict__ off, const float* __restrict__ aw,
    float* __restrict__ out, int Nq) {
  int t = blockIdx.x * blockDim.x + threadIdx.x;
  if (t >= Nq * 8) return;
  int q = t >> 3, h = t & 7;

  const float* awp = aw + (size_t)q * 64 + h * 8;
  float w[8], mx = -1e30f;
#pragma unroll
  for (int p = 0; p < 8; p++) { w[p] = awp[p]; mx = fmaxf(mx, w[p]); }
  float s = 0.f;
#pragma unroll
  for (int p = 0; p < 8; p++) { w[p] = __expf(w[p] - mx); s += w[p]; }
  float inv = 1.f / s;

  float r0 = ref[q * 3 + 0], r1 = ref[q * 3 + 1], r2 = ref[q * 3 + 2];
  const float* offp = off + (size_t)q * 192 + h * 24;

  float acc[16];
#pragma unroll
  for (int d = 0; d < 16; d++) acc[d] = 0.f;

#pragma unroll
  for (int p = 0; p < 8; p++) {
    // coords = (ref + off/shape)*shape - 0.5 = ref*shape + off - 0.5
    float cx = r0 * 128.f + offp[p * 3 + 0] - 0.5f;
    float cy = r1 * 128.f + offp[p * 3 + 1] - 0.5f;
    float cz = r2 * 16.f + offp[p * 3 + 2] - 0.5f;
    float fx = floorf(cx), fy = floorf(cy), fz = floorf(cz);
    float gx = cx - fx, gy = cy - fy, gz = cz - fz;
    int ix = (int)fx, iy = (int)fy, iz = (int)fz;
    float wp = w[p] * inv;
#pragma unroll
    for (int c = 0; c < 8; c++) {
      int dx = c & 1, dy = (c >> 1) & 1, dz = c >> 2;
      int jx = ix + dx, jy = iy + dy, jz = iz + dz;
      if (jx < 0 || jx >= 128 || jy < 0 || jy >= 128 || jz < 0 || jz >= 16) continue;
      float ww = wp * (dx ? gx : 1.f - gx) * (dy ? gy : 1.f - gy) * (dz ? gz : 1.f - gz);
      const float* vp = v + ((size_t)(jx * 2048 + jy * 16 + jz) * 128 + h * 16);
#pragma unroll
      for (int d = 0; d < 16; d++) acc[d] += ww * vp[d];
    }
  }
  float* op = out + (size_t)q * 128 + h * 16;
#pragma unroll
  for (int d = 0; d < 16; d++) op[d] = acc[d];
}

// ---------------------------------------------------------------------------
// Deformable sampling, 2D, 3 levels (48,160)/(24,80)/(12,40), 4 points.
// v: [10080][128], ref: [Nq][2], off: [Nq][192], aw: [Nq][96].
// ---------------------------------------------------------------------------
__global__ __launch_bounds__(256) void deform_sample_2d_kernel(
    const float* __restrict__ v, const float* __restrict__ ref,
    const float* __restrict__ off, const float* __restrict__ aw,
    float* __restrict__ out, int Nq) {
  int t = blockIdx.x * blockDim.x + threadIdx.x;
  if (t >= Nq * 8) return;
  int q = t >> 3, h = t & 7;

  const int Hs[3] = {48, 24, 12};
  const int Ws[3] = {160, 80, 40};
  const int st[3] = {0, 7680, 9600};

  const float* awp = aw + (size_t)q * 96 + h * 12;
  float w[12], mx = -1e30f;
#pragma unroll
  for (int j = 0; j < 12; j++) { w[j] = awp[j]; mx = fmaxf(mx, w[j]); }
  float s = 0.f;
#pragma unroll
  for (int j = 0; j < 12; j++) { w[j] = __expf(w[j] - mx); s += w[j]; }
  float inv = 1.f / s;

  float r0 = ref[q * 2 + 0], r1 = ref[q * 2 + 1];
  const float* offq = off + (size_t)q * 192;

  float acc[16];
#pragma unroll
  for (int d = 0; d < 16; d++) acc[d] = 0.f;

#pragma unroll
  for (int l = 0; l < 3; l++) {
    float Hf = (float)Hs[l], Wf = (float)Ws[l];
#pragma unroll
    for (int p = 0; p < 4; p++) {
      int oi = ((h * 3 + l) * 4 + p) * 2;
      float cy = r0 * Hf + offq[oi + 0] - 0.5f;
      float cx = r1 * Wf + offq[oi + 1] - 0.5f;
      float fy = floorf(cy), fx = floorf(cx);
      float gy = cy - fy, gx = cx - fx;
      int iy = (int)fy, ix = (int)fx;
      float wp = w[l * 4 + p] * inv;
#pragma unroll
      for (int c = 0; c < 4; c++) {
        int dy = c >> 1, dx = c & 1;
        int jy = iy + dy, jx = ix + dx;
        if (jy < 0 || jy >= Hs[l] || jx < 0 || jx >= Ws[l]) continue;
        float ww = wp * (dy ? gy : 1.f - gy) * (dx ? gx : 1.f - gx);
        const float* vp = v + ((size_t)(st[l] + jy * Ws[l] + jx) * 128 + h * 16);
#pragma unroll
        for (int d = 0; d < 16; d++) acc[d] += ww * vp[d];
      }
    }
  }
  float* op = out + (size_t)q * 128 + h * 16;
#pragma unroll
  for (int d = 0; d < 16; d++) op[d] = acc[d];
}

// ---------------------------------------------------------------------------
// Elementwise / gather / scatter helpers
// ---------------------------------------------------------------------------
__global__ void add_vec_kernel(const float* __restrict__ a, const float* __restrict__ b,
                               float* __restrict__ c, int n) {
  int i = blockIdx.x * blockDim.x + threadIdx.x;
  if (i < n) c[i] = a[i] + b[i];
}

// dst[i][c] = src[c*srcStride + idx[i]]  (LDS-tiled transpose-gather)
__global__ void gather_rows_T_kernel(const float* __restrict__ src, const int* __restrict__ idx,
                                     float* __restrict__ dst, int srcStride, int C) {
  __shared__ float tile[32][33];
  int i0 = blockIdx.x * 32, c0 = blockIdx.y * 32;
  int tx = threadIdx.x, ty = threadIdx.y;
  int gi = idx[i0 + tx];
#pragma unroll
  for (int r = 0; r < 4; r++) {
    int c = ty * 4 + r;
    tile[c][tx] = src[(size_t)(c0 + c) * srcStride + gi];
  }
  __syncthreads();
#pragma unroll
  for (int r = 0; r < 4; r++) {
    int i = ty * 4 + r;
    dst[(size_t)(i0 + i) * C + c0 + tx] = tile[tx][i];
  }
}

// dst[c*dstStride + idx[i]] = src[i][c]  (LDS-tiled transpose-scatter)
__global__ void scatter_cols_T_kernel(const float* __restrict__ src, const int* __restrict__ idx,
                                      float* __restrict__ dst, int dstStride, int C) {
  __shared__ float tile[32][33];
  int i0 = blockIdx.x * 32, c0 = blockIdx.y * 32;
  int tx = threadIdx.x, ty = threadIdx.y;
#pragma unroll
  for (int r = 0; r < 4; r++) {
    int i = ty * 4 + r;
    tile[tx][i] = src[(size_t)(i0 + i) * C + c0 + tx];
  }
  __syncthreads();
  int gi = idx[i0 + tx];
#pragma unroll
  for (int r = 0; r < 4; r++) {
    int c = ty * 4 + r;
    dst[(size_t)(c0 + c) * dstStride + gi] = tile[c][tx];
  }
}

// dst[idx[i]*C + c] = src[i*C + c]  (row scatter, fully coalesced per row)
__global__ void scatter_rows_kernel(const float* __restrict__ src, const int* __restrict__ idx,
                                    float* __restrict__ dst, int C) {
  int t = blockIdx.x * blockDim.x + threadIdx.x;
  int i = t >> 7, c = t & 127;
  dst[(size_t)idx[i] * C + c] = src[t];
}

__global__ void gather_ref_kernel(const float* __restrict__ src, const int* __restrict__ idx,
                                  float* __restrict__ dst, int n3) {
  int t = blockIdx.x * blockDim.x + threadIdx.x;
  if (t < n3) {
    int i = t / 3, d = t - i * 3;
    dst[t] = src[(size_t)idx[i] * 3 + d];
  }
}

// ---------------------------------------------------------------------------
// Host-side orchestration
// ---------------------------------------------------------------------------
static inline void launch_gemm(hipStream_t s, const float* A, int lda, int atrans,
                               const float* W, const float* bias, const float* res,
                               float* out, int M, int N, int K, int epi) {
  dim3 grid((M + 63) / 64, (N + 127) / 128);
  size_t sh = (size_t)(64 * AS_PAD + 64 * BS_STRIDE) * sizeof(float);
  gemm_wmma_kernel<<<grid, 256, sh, s>>>(A, lda, atrans, W, bias, res, out, M, N, K, epi);
}

static inline void launch_ln(hipStream_t s, const float* x, const float* g, const float* b,
                             float* out, int M) {
  layernorm_kernel<<<(M + 7) / 8, 256, 0, s>>>(x, g, b, out, M);
}

static inline void launch_add(hipStream_t s, const float* a, const float* b, float* c, int n) {
  add_vec_kernel<<<(n + 255) / 256, 256, 0, s>>>(a, b, c, n);
}

// sorted-key (pytree) order within each param sub-dict:
struct TF {  // b1 b2 bk bo bq bv ln1_b ln1_g ln2_b ln2_g w1 w2 wk wo wq wv
  const float *b1, *b2, *bk, *bo, *bq, *bv, *ln1b, *ln1g, *ln2b, *ln2g, *w1, *w2, *wk, *wo, *wq, *wv;
};
struct DF {  // b1 b2 b_aw b_o b_off b_v ln1_b ln1_g ln2_b ln2_g w1 w2 w_aw w_o w_off w_v
  const float *b1, *b2, *baw, *bo, *boff, *bv, *ln1b, *ln1g, *ln2b, *ln2g, *w1, *w2, *waw, *wo, *woff, *wv;
};

static inline TF tf_params(void* const* din, int b) {
  const float* p[16];
  for (int i = 0; i < 16; i++) p[i] = (const float*)din[b + i];
  return TF{p[0], p[1], p[2], p[3], p[4], p[5], p[6], p[7], p[8], p[9], p[10], p[11], p[12], p[13], p[14], p[15]};
}
static inline DF df_params(void* const* din, int b) {
  const float* p[16];
  for (int i = 0; i < 16; i++) p[i] = (const float*)din[b + i];
  return DF{p[0], p[1], p[2], p[3], p[4], p[5], p[6], p[7], p[8], p[9], p[10], p[11], p[12], p[13], p[14], p[15]};
}

static inline void run_ffn(hipStream_t s, float* x, float* t, float* h, float* x2, int M,
                           const float* ln1g, const float* ln1b, const float* w1, const float* b1,
                           const float* w2, const float* b2, const float* ln2g, const float* ln2b) {
  launch_ln(s, x, ln1g, ln1b, t, M);
  launch_gemm(s, t, 128, 0, w1, b1, nullptr, h, M, 512, 128, 1);  // relu
  launch_gemm(s, h, 512, 0, w2, b2, t, x2, M, 128, 512, 2);       // +t residual
  launch_ln(s, x2, ln2g, ln2b, x, M);
}

extern "C" void kernel_launch(void* const* d_in, const int* in_sizes, int n_in,
                              void* d_out, int out_size, void* d_ws, size_t ws_size,
                              hipStream_t stream) {
  (void)in_sizes; (void)n_in; (void)out_size; (void)ws_size;

  // ---- inputs (setup_inputs order; params pytree-flattened alphabetically) --
  const float* scene_embed = (const float*)d_in[0];  // [128][NVOX] channel-major
  const float* inst_queries = (const float*)d_in[1]; // [100][128]
  const float* feat0 = (const float*)d_in[2];        // [128][7680]
  const float* feat1 = (const float*)d_in[3];        // [128][1920]
  const float* feat2 = (const float*)d_in[4];        // [128][480]
  const float* scene_pos = (const float*)d_in[5];    // [128][NVOX]
  const float* inst_pos = (const float*)d_in[6];     // [100][128]
  const float* ref_2d = (const float*)d_in[7];       // [100][2]
  const float* ref_3d = (const float*)d_in[8];       // [100][3]
  const float* ref_vox = (const float*)d_in[9];      // [NVOX][3]
  const int* fidx = (const int*)d_in[10];            // [NFOV]
  DF qida = df_params(d_in, 11);
  TF qsa = tf_params(d_in, 27);
  DF qsda = df_params(d_in, 43);
  TF sqca = tf_params(d_in, 59);
  DF ssda = df_params(d_in, 75);

  // ---- workspace layout ----------------------------------------------------
  float* ws = (float*)d_ws;
  size_t off = 0;
  auto take = [&](size_t nf) { float* r = ws + off; off += (nf + 63) & ~(size_t)63; return r; };
  float* v_big = take((size_t)NVOX * 128);       // ssda value, then qsda value
  float* fov_h = take((size_t)NFOV * 512);       // FFN hidden; aliased below
  float* f_off = fov_h;                          // [NFOV][192] (alias, disjoint in time)
  float* f_aw = fov_h + (size_t)NFOV * 192;      // [NFOV][64]
  float* fov_x = take((size_t)NFOV * 128);
  float* fov_t = take((size_t)NFOV * 128);
  float* fov_x2 = take((size_t)NFOV * 128);
  float* f_qh = take((size_t)NFOV * 128);
  float* f_tmp = f_qh;                           // alias: qsda fov value proj
  float* f_att = take((size_t)NFOV * 128);
  float* f_samp = f_att;                         // alias: attn out consumed first
  float* f_qin = take((size_t)NFOV * 128);
  float* pos_fov = take((size_t)NFOV * 128);
  float* ref_fov = take((size_t)NFOV * 4);
  float* v_feat = take((size_t)10080 * 128);
  float* i_x = take((size_t)NINST * 128);
  float* i_t = take((size_t)NINST * 128);
  float* i_x2 = take((size_t)NINST * 128);
  float* i_h = take((size_t)NINST * 512);
  float* i_qin = take((size_t)NINST * 128);
  float* i_kin = take((size_t)NINST * 128);
  float* i_qh = take((size_t)NINST * 128);
  float* i_kh = take((size_t)NINST * 128);
  float* i_vh = take((size_t)NINST * 128);
  float* i_att = take((size_t)NINST * 128);
  float* i_samp = take((size_t)NINST * 128);
  float* i_off = take((size_t)NINST * 192);
  float* i_aw = take((size_t)NINST * 96);

  float* out_scene = (float*)d_out;                      // [128][NVOX]
  float* out_inst = (float*)d_out + (size_t)NVOX * 128;  // [100][128]

  // Scene output starts as a copy of scene_embed (fov columns overwritten later)
  (void)hipMemcpyAsync(out_scene, scene_embed, (size_t)NVOX * 128 * sizeof(float),
                       hipMemcpyDeviceToDevice, stream);

  // ==== Stage A: qsa self-attention on instance queries =====================
  launch_add(stream, inst_queries, inst_pos, i_qin, NINST * 128);
  launch_gemm(stream, i_qin, 128, 0, qsa.wq, qsa.bq, nullptr, i_qh, NINST, 128, 128, 0);
  launch_gemm(stream, i_qin, 128, 0, qsa.wk, qsa.bk, nullptr, i_kh, NINST, 128, 128, 0);
  launch_gemm(stream, inst_queries, 128, 0, qsa.wv, qsa.bv, nullptr, i_vh, NINST, 128, 128, 0);
  mha_attn_kernel<<<NINST, 256, 0, stream>>>(i_qh, i_kh, i_vh, i_att, NINST, NINST);
  launch_gemm(stream, i_att, 128, 0, qsa.wo, qsa.bo, inst_queries, i_x, NINST, 128, 128, 2);
  run_ffn(stream, i_x, i_t, i_h, i_x2, NINST, qsa.ln1g, qsa.ln1b, qsa.w1, qsa.b1,
          qsa.w2, qsa.b2, qsa.ln2g, qsa.ln2b);

  // ==== Stage B: gather FOV tokens =========================================
  {
    dim3 g(NFOV / 32, 128 / 32), b(32, 8);
    gather_rows_T_kernel<<<g, b, 0, stream>>>(scene_embed, fidx, fov_x, NVOX, 128);
    gather_rows_T_kernel<<<g, b, 0, stream>>>(scene_pos, fidx, pos_fov, NVOX, 128);
  }
  gather_ref_kernel<<<(NFOV * 3 + 255) / 256, 256, 0, stream>>>(ref_vox, fidx, ref_fov, NFOV * 3);

  // ==== Stage C: sqca cross-attention (scene_fov x inst_q) ==================
  launch_add(stream, fov_x, pos_fov, f_qin, NFOV * 128);
  launch_add(stream, i_x, inst_pos, i_kin, NINST * 128);
  launch_gemm(stream, f_qin, 128, 0, sqca.wq, sqca.bq, nullptr, f_qh, NFOV, 128, 128, 0);
  launch_gemm(stream, i_kin, 128, 0, sqca.wk, sqca.bk, nullptr, i_kh, NINST, 128, 128, 0);
  launch_gemm(stream, i_x, 128, 0, sqca.wv, sqca.bv, nullptr, i_vh, NINST, 128, 128, 0);
  mha_attn_kernel<<<NFOV, 256, 0, stream>>>(f_qh, i_kh, i_vh, f_att, NFOV, NINST);
  launch_gemm(stream, f_att, 128, 0, sqca.wo, sqca.bo, fov_x, fov_x, NFOV, 128, 128, 2);
  run_ffn(stream, fov_x, fov_t, fov_h, fov_x2, NFOV, sqca.ln1g, sqca.ln1b, sqca.w1, sqca.b1,
          sqca.w2, sqca.b2, sqca.ln2g, sqca.ln2b);

  // ==== Stage D: ssda deformable attention over full scene ==================
  // value proj directly from channel-major scene_embed (A-transposed GEMM)
  launch_gemm(stream, scene_embed, NVOX, 1, ssda.wv, ssda.bv, nullptr, v_big, NVOX, 128, 128, 0);
  launch_add(stream, fov_x, pos_fov, f_qin, NFOV * 128);
  launch_gemm(stream, f_qin, 128, 0, ssda.woff, ssda.boff, nullptr, f_off, NFOV, 192, 128, 0);
  launch_gemm(stream, f_qin, 128, 0, ssda.waw, ssda.baw, nullptr, f_aw, NFOV, 64, 128, 0);
  deform_sample_3d_kernel<<<(NFOV * 8 + 255) / 256, 256, 0, stream>>>(
      v_big, ref_fov, f_off, f_aw, f_samp, NFOV);
  launch_gemm(stream, f_samp, 128, 0, ssda.wo, ssda.bo, fov_x, fov_x, NFOV, 128, 128, 2);
  run_ffn(stream, fov_x, fov_t, fov_h, fov_x2, NFOV, ssda.ln1g, ssda.ln1b, ssda.w1, ssda.b1,
          ssda.w2, ssda.b2, ssda.ln2g, ssda.ln2b);

  // ==== Stage E: scene output scatter ======================================
  {
    dim3 g(NFOV / 32, 128 / 32), b(32, 8);
    scatter_cols_T_kernel<<<g, b, 0, stream>>>(fov_x, fidx, out_scene, NVOX, 128);
  }

  // ==== Stage F: qsda deformable attention (inst over updated scene) =======
  launch_gemm(stream, scene_embed, NVOX, 1, qsda.wv, qsda.bv, nullptr, v_big, NVOX, 128, 128, 0);
  launch_gemm(stream, fov_x, 128, 0, qsda.wv, qsda.bv, nullptr, f_tmp, NFOV, 128, 128, 0);
  scatter_rows_kernel<<<(NFOV * 128) / 256, 256, 0, stream>>>(f_tmp, fidx, v_big, 128);
  launch_add(stream, i_x, inst_pos, i_qin, NINST * 128);
  launch_gemm(stream, i_qin, 128, 0, qsda.woff, qsda.boff, nullptr, i_off, NINST, 192, 128, 0);
  launch_gemm(stream, i_qin, 128, 0, qsda.waw, qsda.baw, nullptr, i_aw, NINST, 64, 128, 0);
  deform_sample_3d_kernel<<<(NINST * 8 + 255) / 256, 256, 0, stream>>>(
      v_big, ref_3d, i_off, i_aw, i_samp, NINST);
  launch_gemm(stream, i_samp, 128, 0, qsda.wo, qsda.bo, i_x, i_x, NINST, 128, 128, 2);
  run_ffn(stream, i_x, i_t, i_h, i_x2, NINST, qsda.ln1g, qsda.ln1b, qsda.w1, qsda.b1,
          qsda.w2, qsda.b2, qsda.ln2g, qsda.ln2b);

  // ==== Stage G: qida deformable attention over image features ==============
  launch_gemm(stream, feat0, 7680, 1, qida.wv, qida.bv, nullptr, v_feat, 7680, 128, 128, 0);
  launch_gemm(stream, feat1, 1920, 1, qida.wv, qida.bv, nullptr, v_feat + (size_t)7680 * 128, 1920, 128, 128, 0);
  launch_gemm(stream, feat2, 480, 1, qida.wv, qida.bv, nullptr, v_feat + (size_t)9600 * 128, 480, 128, 128, 0);
  launch_add(stream, i_x, inst_pos, i_qin, NINST * 128);
  launch_gemm(stream, i_qin, 128, 0, qida.woff, qida.boff, nullptr, i_off, NINST, 192, 128, 0);
  launch_gemm(stream, i_qin, 128, 0, qida.waw, qida.baw, nullptr, i_aw, NINST, 96, 128, 0);
  deform_sample_2d_kernel<<<(NINST * 8 + 255) / 256, 256, 0, stream>>>(
      v_feat, ref_2d, i_off, i_aw, i_samp, NINST);
  launch_gemm(stream, i_samp, 128, 0, qida.wo, qida.bo, i_x, i_x, NINST, 128, 128, 2);
  run_ffn(stream, i_x, i_t, i_h, i_x2, NINST, qida.ln1g, qida.ln1b, qida.w1, qida.b1,
          qida.w2, qida.b2, qida.ln2g, qida.ln2b);

  (void)hipMemcpyAsync(out_inst, i_x, (size_t)NINST * 128 * sizeof(float),
                       hipMemcpyDeviceToDevice, stream);
}